// AsphericSurface_14482629722966
// MI455X (gfx1250) — compile-verified
//
#include <hip/hip_runtime.h>
#include <math.h>

#define TILE 256
#define MAX_ITER 10

#define HAS_ASYNC_LDS __has_builtin(__builtin_amdgcn_global_load_async_to_lds_b32)

typedef __attribute__((address_space(1))) int g_as_int;   // global (AS1)
typedef __attribute__((address_space(3))) int s_as_int;   // LDS (AS3)

__device__ __forceinline__ float fast_rcp(float x)  { return __builtin_amdgcn_rcpf(x); }
__device__ __forceinline__ float fast_sqrt(float x) { return __builtin_amdgcn_sqrtf(x); }
__device__ __forceinline__ float fast_rsq(float x)  { return __builtin_amdgcn_rsqf(x); }

struct Surf {
    float c;       // curvature
    float A;       // (1+conic)*c*c
    float cOverA;  // c/A (valid only when useA)
    int   useA;    // A != 0 (wave-uniform)
    float x0, y0, z0;
    float a0, a1, a2, a3;
};

// sag_raw(X, Y) = c*r2/(1+sqrt(1-A*r2)) + a0*r2^2 + a1*r2^4 + a2*r2^6 + a3*r2^8
// (exact-division form; used once for z0)
__device__ __forceinline__ float sag_raw_exact(float X, float Y, const Surf& s) {
    float r2   = fmaf(X, X, Y * Y);
    float sq   = fast_sqrt(fmaf(-s.A, r2, 1.0f));
    float zc   = r2 * s.c * fast_rcp(1.0f + sq);
    float r4   = r2 * r2;
    float poly = r4 * fmaf(r4, fmaf(r4, fmaf(r4, s.a3, s.a2), s.a1), s.a0);
    return zc + poly;
}

// g(t) = (oz + z0) + t*dz - sag_raw(ox+x0 + t*dx, oy+y0 + t*dy)
// g'(t) = dz - dzdr2 * dr2dt, dzdr2 = c/(2*sqrt(1-A*r2)) + poly'(r2)
// conic sag uses r2/(1+s) == (1-s)/A  (exact limit 0.5*r2 when A==0)
__device__ __forceinline__ void eval_g(float t,
                                       float oxs, float oys, float ozs,
                                       float dx, float dy, float dz,
                                       const Surf& s, float& g, float& gp) {
    float X      = fmaf(t, dx, oxs);
    float Y      = fmaf(t, dy, oys);
    float r2     = fmaf(X, X, Y * Y);
    float s2     = fmaf(-s.A, r2, 1.0f);
    float inv_sq = fast_rsq(s2);             // 1/sqrt(s2)
    float sq     = s2 * inv_sq;              // sqrt(s2)
    float zc     = s.useA ? (s.cOverA * (1.0f - sq)) : (0.5f * s.c * r2);
    float r4     = r2 * r2;
    float poly   = r4 * fmaf(r4, fmaf(r4, fmaf(r4, s.a3, s.a2), s.a1), s.a0);
    float Z      = fmaf(t, dz, ozs);
    g = Z - (zc + poly);

    float dpoly  = r2 * fmaf(r4, fmaf(r4, fmaf(r4, 8.0f * s.a3, 6.0f * s.a2),
                                      4.0f * s.a1), 2.0f * s.a0);
    float dzdr2  = fmaf(0.5f * s.c, inv_sq, dpoly);
    float dr2dt  = 2.0f * fmaf(X, dx, Y * dy);
    gp = fmaf(-dzdr2, dr2dt, dz);
}

__global__ __launch_bounds__(TILE)
void aspheric_newton_kernel(const float* __restrict__ origin,
                            const float* __restrict__ dir,
                            const float* __restrict__ offset,
                            const float* __restrict__ curvature,
                            const float* __restrict__ conic,
                            const float* __restrict__ aspheric,
                            float* __restrict__ out, int N) {
    const int tid = threadIdx.x;
    const int ray = blockIdx.x * TILE + tid;

    __shared__ float tO[3 * TILE];
    __shared__ float tD[3 * TILE];

    const int fmax  = 3 * N - 1;              // 3N < 2^31 for this workload
    const int fbase = blockIdx.x * (3 * TILE);

#if HAS_ASYNC_LDS
    // CDNA5 async global -> LDS staging (tracked by ASYNCcnt)
    #pragma unroll
    for (int j = 0; j < 3; ++j) {
        int fi = fbase + j * TILE + tid;
        if (fi > fmax) fi = fmax;
        __builtin_amdgcn_global_load_async_to_lds_b32(
            (g_as_int*)(origin + fi),
            (s_as_int*)(&tO[j * TILE + tid]),
            0, 0);
        __builtin_amdgcn_global_load_async_to_lds_b32(
            (g_as_int*)(dir + fi),
            (s_as_int*)(&tD[j * TILE + tid]),
            0, 0);
    }
#if __has_builtin(__builtin_amdgcn_s_wait_asynccnt)
    __builtin_amdgcn_s_wait_asynccnt(0);
#else
    asm volatile("s_wait_asynccnt 0" ::: "memory");
#endif
#else
    #pragma unroll
    for (int j = 0; j < 3; ++j) {
        int fi = fbase + j * TILE + tid;
        if (fi > fmax) fi = fmax;
        tO[j * TILE + tid] = origin[fi];
        tD[j * TILE + tid] = dir[fi];
    }
#endif
    __syncthreads();

    // Uniform surface constants (scalarized by compiler: s_load)
    Surf s;
    s.c = curvature[0];
    float k = conic[0];
    s.A      = (1.0f + k) * s.c * s.c;
    s.useA   = (s.A != 0.0f);
    s.cOverA = s.c * fast_rcp(s.A);          // only consumed when useA
    s.x0 = offset[0];
    s.y0 = offset[1];
    s.a0 = aspheric[0];
    s.a1 = aspheric[1];
    s.a2 = aspheric[2];
    s.a3 = aspheric[3];
    s.z0 = sag_raw_exact(s.x0, s.y0, s);

    if (ray >= N) return;

    float ox = tO[3 * tid + 0], oy = tO[3 * tid + 1], oz = tO[3 * tid + 2];
    float dx = tD[3 * tid + 0], dy = tD[3 * tid + 1], dz = tD[3 * tid + 2];

    // t_init = max(0, -oz/dz) when |dz| > 1e-10 else 0
    float t = (fabsf(dz) > 1e-10f) ? (-oz * fast_rcp(dz)) : 0.0f;
    t = fmaxf(t, 0.0f);

    const float oxs = ox + s.x0;
    const float oys = oy + s.y0;
    const float ozs = oz + s.z0;

    bool conv = false;
    for (int it = 0; it < MAX_ITER; ++it) {
        float g, gp;
        eval_g(t, oxs, oys, ozs, dx, dy, dz, s, g, gp);
        float gps  = (fabsf(gp) > 1e-12f) ? gp : 1e-12f;
        float tn   = t - g * fast_rcp(gps);
        bool  cnew = conv || (fabsf(g) < 1e-5f);
        t    = conv ? t : tn;      // gated on *previous* converged flag (matches scan)
        conv = cnew;
#if __has_builtin(__builtin_amdgcn_ballot_w32)
        if (__builtin_amdgcn_ballot_w32(!conv) == 0u) break;  // wave-uniform early exit
#endif
    }

    // Final hit point, residual, normal
    float x_hit = fmaf(t, dx, ox);
    float y_hit = fmaf(t, dy, oy);
    float X = x_hit + s.x0;
    float Y = y_hit + s.y0;

    float r2     = fmaf(X, X, Y * Y);
    float s2     = fmaf(-s.A, r2, 1.0f);
    float inv_sq = fast_rsq(s2);
    float sq     = s2 * inv_sq;
    float zc     = s.useA ? (s.cOverA * (1.0f - sq)) : (0.5f * s.c * r2);
    float r4     = r2 * r2;
    float poly   = r4 * fmaf(r4, fmaf(r4, fmaf(r4, s.a3, s.a2), s.a1), s.a0);
    float sagv   = zc + poly;            // sag_raw(X, Y)
    float z_hit  = sagv - s.z0;
    float g_res  = fmaf(t, dz, oz) - z_hit;

    float dpoly  = r2 * fmaf(r4, fmaf(r4, fmaf(r4, 8.0f * s.a3, 6.0f * s.a2),
                                      4.0f * s.a1), 2.0f * s.a0);
    float dzdr2  = fmaf(0.5f * s.c, inv_sq, dpoly);
    float dzdx   = 2.0f * X * dzdr2;
    float dzdy   = 2.0f * Y * dzdr2;
    float inv_n  = fast_rsq(fmaf(dzdx, dzdx, fmaf(dzdy, dzdy, 1.0f)));
    float nx = -dzdx * inv_n;
    float ny = -dzdy * inv_n;
    float nz = inv_n;

    bool  valid = (t > 1e-8f) && (fabsf(g_res) < 1e-3f);
    float t_out = valid ? t : __builtin_inff();

    // Output layout: [0,N) t | [N,4N) point | [4N,7N) normal
    out[ray] = t_out;
    size_t pb = (size_t)N + 3u * (size_t)ray;
    out[pb + 0] = x_hit;
    out[pb + 1] = y_hit;
    out[pb + 2] = z_hit;
    size_t nb = 4u * (size_t)N + 3u * (size_t)ray;
    out[nb + 0] = nx;
    out[nb + 1] = ny;
    out[nb + 2] = nz;
}

extern "C" void kernel_launch(void* const* d_in, const int* in_sizes, int n_in,
                              void* d_out, int out_size, void* d_ws, size_t ws_size,
                              hipStream_t stream) {
    const float* origin    = (const float*)d_in[0];
    const float* dir       = (const float*)d_in[1];
    const float* offset    = (const float*)d_in[2];
    const float* curvature = (const float*)d_in[3];
    const float* conic     = (const float*)d_in[4];
    const float* aspheric  = (const float*)d_in[5];
    float*       out       = (float*)d_out;

    const int N      = in_sizes[0] / 3;
    const int blocks = (N + TILE - 1) / TILE;

    aspheric_newton_kernel<<<blocks, TILE, 0, stream>>>(
        origin, dir, offset, curvature, conic, aspheric, out, N);
}